// SADPixelwise_27255862460502
// MI455X (gfx1250) — compile-verified
//
#include <hip/hip_runtime.h>
#include <hip/hip_bf16.h>
#include <math.h>

// ---------------------------------------------------------------------------
// SAM (spectral angle) loss, MI455X / gfx1250, wave32.
//
// input:     (8, 31, 512, 512) f32   band-major
// target:    (8, 1, 512, 512)  i32   material code 0..15 (255 = ignore)
// materials: (16, 31)          f32
// out:       scalar f32 = sum over valid pixels of arccos(cos_angle)
//
// D(16pix x 16mat) = X_tile(16x32, band31=0) x M^T(32x16) via 8x
// V_WMMA_F32_16X16X4_F32 (f32 precision matches reference). Two tiles are
// paired so the bpermute-gathered per-pixel dot/norm/code fill all 32 lanes
// and the expensive div/sqrt/acos epilogue runs once per 32 pixels at full
// lane occupancy. Streamed arrays use non-temporal loads (no reuse, keep L2
// clean). Roofline: ~268 MB => ~11.5 us @ 23.3 TB/s.
// ---------------------------------------------------------------------------

typedef __attribute__((ext_vector_type(2))) float v2f;
typedef __attribute__((ext_vector_type(8))) float v8f;

#define N_BANDS 31
#define N_MATS  16
#define IMG_HW  (512 * 512)
#define N_PIX   (8 * IMG_HW)
#define N_TILES (N_PIX / 16)
#define N_PAIRS (N_TILES / 2)

#define GRID_BLOCKS 1024
#define BLOCK_THREADS 256

// Process one 16-pixel tile: 8 WMMAs + ||x||^2, then gather the code-selected
// dot so that lane L holds pixel (L%16)'s {dot, ||x||^2, code}.
__device__ __forceinline__ void tile_dotsel(const float* __restrict__ xb,
                                            const int*   __restrict__ tb,
                                            const float bm[8][2],
                                            const int boff[8][2],
                                            int lane, int half, int myc,
                                            float& dotSel, float& xn2, int& code)
{
    // codes: lane L loads code of pixel L%16 (halves duplicate)
    code = __builtin_nontemporal_load(tb + myc);

    v8f c = {};        // 16x16 f32 accumulator: pixels x materials
    float xn = 0.0f;   // per-lane partial ||x||^2 for pixel myc

    // A layout (16x4 f32): VGPR j, lane L -> M = L%16, K = kk*4 + j + 2*half
#pragma unroll
    for (int kk = 0; kk < 8; ++kk) {
        v2f a;
#pragma unroll
        for (int j = 0; j < 2; ++j) {
            int band = kk * 4 + j + 2 * half;
            float v = __builtin_nontemporal_load(xb + boff[kk][j]);
            v = (band < N_BANDS) ? v : 0.0f;   // zero-pad band 31
            a[j] = v;
            xn = fmaf(v, v, xn);
        }
        v2f b;
        b[0] = bm[kk][0];
        b[1] = bm[kk][1];
        // D = A x B + C ; 8 args: (neg_a, A, neg_b, B, c_mod, C, reuse_a, reuse_b)
        c = __builtin_amdgcn_wmma_f32_16x16x4_f32(
                false, a, false, b, (short)0, c, false, false);
    }

    // pixel p's ||x||^2 is split across lanes p and p+16 -> after the merge
    // lane L holds pixel (L%16)'s total.
    xn2 = xn + __shfl_xor(xn, 16, 32);

    // Row r's selected dot lives in lane code[r] + 16*(r>=8), VGPR r%8.
    // Gather it to lane L (r = L%16): 8 bpermutes + 8 selects.
    const int srcLane = (code & 31) + ((myc >= 8) ? 16 : 0);
    float d = 0.0f;
#pragma unroll
    for (int v = 0; v < 8; ++v) {
        float t = __shfl(c[v], srcLane, 32);
        d = ((lane & 7) == v) ? t : d;
    }
    dotSel = d;
}

__global__ __launch_bounds__(BLOCK_THREADS)
void sam_wmma_kernel(const float* __restrict__ x,
                     const int*   __restrict__ target,
                     const float* __restrict__ mats,
                     float*       __restrict__ partial)
{
    const int tid         = threadIdx.x;
    const int lane        = tid & 31;
    const int waveInBlock = tid >> 5;
    const int wavesPerBlk = BLOCK_THREADS >> 5;
    const int waveGlobal  = blockIdx.x * wavesPerBlk + waveInBlock;
    const int totalWaves  = GRID_BLOCKS * wavesPerBlk;

    const int half = lane >> 4;   // 0: lanes 0-15, 1: lanes 16-31
    const int myc  = lane & 15;   // material column (C) AND pixel row (A)

    // ---- loop-invariant B fragments (materials) and per-lane load offsets.
    //      B layout (4x16 f32): VGPR j, lane L -> K = j + 2*half, N = L%16
    float bm[8][2];
    int   boff[8][2];
#pragma unroll
    for (int kk = 0; kk < 8; ++kk) {
#pragma unroll
        for (int j = 0; j < 2; ++j) {
            int band  = kk * 4 + j + 2 * half;
            int sband = band < N_BANDS ? band : 0;
            float v = mats[myc * N_BANDS + sband];
            bm[kk][j]   = (band < N_BANDS) ? v : 0.0f;
            boff[kk][j] = sband * IMG_HW + myc;    // constant across tiles
        }
    }

    // ---- material norm^2: lane L holds tn2 of material L%16
    float tn2 = 0.0f;
    for (int b = 0; b < N_BANDS; ++b) {
        float m = mats[myc * N_BANDS + b];
        tn2 = fmaf(m, m, tn2);
    }

    float acc = 0.0f;

    for (int pair = waveGlobal; pair < N_PAIRS; pair += totalWaves) {
        const int pix0 = pair * 32;              // 32 consecutive pixels
        const int img  = pix0 / IMG_HW;          // IMG_HW % 32 == 0: no straddle
        const int pp   = pix0 - img * IMG_HW;
        const float* xb = x + (size_t)img * N_BANDS * IMG_HW + pp;
        const int*   tb = target + (size_t)img * IMG_HW + pp;

        float dA, dB, xA, xB;
        int   cA, cB;
        tile_dotsel(xb,      tb,      bm, boff, lane, half, myc, dA, xA, cA);
        tile_dotsel(xb + 16, tb + 16, bm, boff, lane, half, myc, dB, xB, cB);

        // lanes 0-15 handle tile A pixels 0-15; lanes 16-31 tile B pixels 0-15
        float dot = half ? dB : dA;
        float xn2 = half ? xB : xA;
        int   cd  = half ? cB : cA;

        float tg    = __shfl(tn2, cd & 15, 32);  // tn2[material]
        int   valid = (cd >= 0) & (cd < N_MATS); // 255 (ignore) fails

        float cosv = dot / sqrtf(xn2 * tg);
        cosv = fminf(fmaxf(cosv, -1.0f + 1e-7f), 1.0f - 1e-7f);
        float ang = acosf(cosv);
        acc += valid ? ang : 0.0f;
    }

    // ---- deterministic wave reduction
#pragma unroll
    for (int off = 16; off >= 1; off >>= 1)
        acc += __shfl_xor(acc, off, 32);

    __shared__ float smem[BLOCK_THREADS / 32];
    if (lane == 0) smem[waveInBlock] = acc;
    __syncthreads();
    if (tid == 0) {
        float s = 0.0f;
        for (int w = 0; w < wavesPerBlk; ++w) s += smem[w];
        partial[blockIdx.x] = s;
    }
}

// Fixed-order final reduction: deterministic across replays.
__global__ __launch_bounds__(BLOCK_THREADS)
void sam_reduce_kernel(const float* __restrict__ partial, int n,
                       float* __restrict__ out)
{
    __shared__ float sh[BLOCK_THREADS];
    float s = 0.0f;
    for (int i = threadIdx.x; i < n; i += BLOCK_THREADS) s += partial[i];
    sh[threadIdx.x] = s;
    __syncthreads();
#pragma unroll
    for (int off = BLOCK_THREADS / 2; off >= 1; off >>= 1) {
        if ((int)threadIdx.x < off) sh[threadIdx.x] += sh[threadIdx.x + off];
        __syncthreads();
    }
    if (threadIdx.x == 0) out[0] = sh[0];
}

extern "C" void kernel_launch(void* const* d_in, const int* in_sizes, int n_in,
                              void* d_out, int out_size, void* d_ws, size_t ws_size,
                              hipStream_t stream)
{
    const float* x      = (const float*)d_in[0];  // (8,31,512,512) f32
    const int*   target = (const int*)  d_in[1];  // (8,1,512,512)  i32
    const float* mats   = (const float*)d_in[2];  // (16,31)        f32
    float* out     = (float*)d_out;
    float* partial = (float*)d_ws;                // GRID_BLOCKS floats (4 KB)

    sam_wmma_kernel<<<GRID_BLOCKS, BLOCK_THREADS, 0, stream>>>(x, target, mats, partial);
    sam_reduce_kernel<<<1, BLOCK_THREADS, 0, stream>>>(partial, GRID_BLOCKS, out);
}